// CrowdsClassificationSModel_80650895884838
// MI455X (gfx1250) — compile-verified
//
#include <hip/hip_runtime.h>

typedef float v2f __attribute__((ext_vector_type(2)));
typedef float v8f __attribute__((ext_vector_type(8)));

#define C_DIM   8
#define R_DIM   59
#define NPAIR   30          // ceil(59/2); pair 29 has a zero-padded dummy r=59 column
#define TILE_B  16
#define THREADS 256
#define LDS_STRIDE 480      // 472 + 8 pad floats (both divisible by 4)

// --- Prologue: softmax over d (axis=1) of kernel[c,d,r], packed straight into
// the WMMA B-matrix lane layout:  Bws[(p*8 + k)*16 + n], k=c, n = d + 8*(r&1), p=r>>1.
__global__ void softmax_pack_kernel(const float* __restrict__ kern,
                                    float* __restrict__ Bws) {
    int tid = threadIdx.x;
    // zero everything (covers the dummy r=59 columns of pair 29)
    for (int i = tid; i < NPAIR * 8 * 16; i += blockDim.x) Bws[i] = 0.0f;
    __syncthreads();
    if (tid < C_DIM * R_DIM) {
        int c = tid / R_DIM, r = tid % R_DIM;
        const float* row = kern + c * (C_DIM * R_DIM) + r;   // stride 59 over d
        float v[C_DIM];
        float mx = -INFINITY;
        #pragma unroll
        for (int d = 0; d < C_DIM; ++d) { v[d] = row[d * R_DIM]; mx = fmaxf(mx, v[d]); }
        float s = 0.0f;
        #pragma unroll
        for (int d = 0; d < C_DIM; ++d) { v[d] = __expf(v[d] - mx); s += v[d]; }
        float inv = 1.0f / s;
        int p = r >> 1;
        int hi = (r & 1) * 8;
        #pragma unroll
        for (int d = 0; d < C_DIM; ++d)
            Bws[(p * 8 + c) * 16 + d + hi] = v[d] * inv;
    }
}

// --- Main kernel: one 16-batch-row tile per block; 8 waves split the 30
// annotator-pairs; two chained V_WMMA_F32_16X16X4_F32 per pair (K=8);
// D scattered to LDS; vectorized, branchless, coalesced masked epilogue.
__global__ void __launch_bounds__(THREADS)
crowds_wmma_kernel(const float* __restrict__ x,
                   const float* __restrict__ Bws,
                   const unsigned char* __restrict__ mask,
                   float* __restrict__ out) {
    __shared__ float tile[TILE_B * LDS_STRIDE];

    const int tid  = threadIdx.x;
    const int wave = tid >> 5;
    const int lane = tid & 31;
    const int b0   = blockIdx.x * TILE_B;

    // A fragment (ISA 16x4 f32 layout): lane<16 -> K=0,1 ; lane>=16 -> K=2,3.
    // Shared across every pair this wave processes.
    const int row = lane & 15;
    const int kb  = (lane >> 4) << 1;            // 0 or 2
    const float* xr = x + (size_t)(b0 + row) * C_DIM;
    v2f a_lo, a_hi;
    a_lo[0] = xr[kb];     a_lo[1] = xr[kb + 1];  // K = kb,   kb+1
    a_hi[0] = xr[kb + 4]; a_hi[1] = xr[kb + 5];  // K = kb+4, kb+5

    const int n = lane & 15;                     // N column (d + 8*(r&1))
    for (int p = wave; p < NPAIR; p += (THREADS / 32)) {
        const float* Bp = Bws + p * (8 * 16);
        v2f b_lo, b_hi;
        b_lo[0] = Bp[(kb    ) * 16 + n];
        b_lo[1] = Bp[(kb + 1) * 16 + n];
        b_hi[0] = Bp[(kb + 4) * 16 + n];
        b_hi[1] = Bp[(kb + 5) * 16 + n];

        v8f acc = {};
        acc = __builtin_amdgcn_wmma_f32_16x16x4_f32(
                  false, a_lo, false, b_lo, (short)0, acc, false, false);
        acc = __builtin_amdgcn_wmma_f32_16x16x4_f32(
                  false, a_hi, false, b_hi, (short)0, acc, false, false);

        // D layout: VGPR j holds M=j (lanes 0-15) / M=j+8 (lanes 16-31), N=lane&15
        int d = n & 7;
        int r = 2 * p + (n >> 3);
        if (r < R_DIM) {                         // drop dummy r=59 column
            int col   = d * R_DIM + r;
            int rbase = (lane >> 4) * 8;
            #pragma unroll
            for (int j = 0; j < 8; ++j)
                tile[(rbase + j) * LDS_STRIDE + col] = acc[j];
        }
    }
    __syncthreads();

    // Vectorized (x4) branchless dropout + store. The whole per-block output
    // region is contiguous: 16 rows * 472 floats = 1888 float4s, and
    // 472 % 4 == 0 so a float4 never crosses a row boundary.
    const float inv_keep = 1.0f / 0.6f;
    const int ROW_Q = (C_DIM * R_DIM) / 4;                 // 118 float4 per row
    const uchar4* __restrict__ m4p = (const uchar4*)mask;
    float4*       __restrict__ o4p = (float4*)out;
    const size_t gbase = (size_t)b0 * ROW_Q;               // float4 / uchar4 units

    for (int i = tid; i < TILE_B * ROW_Q; i += THREADS) {
        int b  = i / ROW_Q;
        int kq = i - b * ROW_Q;
        const float4 t = *(const float4*)&tile[b * LDS_STRIDE + kq * 4];
        const uchar4 m = m4p[gbase + i];
        float4 o;
        o.x = t.x * (m.x ? inv_keep : 0.0f);
        o.y = t.y * (m.y ? inv_keep : 0.0f);
        o.z = t.z * (m.z ? inv_keep : 0.0f);
        o.w = t.w * (m.w ? inv_keep : 0.0f);
        o4p[gbase + i] = o;
    }
}

extern "C" void kernel_launch(void* const* d_in, const int* in_sizes, int n_in,
                              void* d_out, int out_size, void* d_ws, size_t ws_size,
                              hipStream_t stream) {
    const float*         x    = (const float*)d_in[0];
    const float*         kern = (const float*)d_in[1];
    const unsigned char* mask = (const unsigned char*)d_in[2];   // jax bool -> bytes
    float* out = (float*)d_out;
    float* Bws = (float*)d_ws;                                   // 3840 floats

    softmax_pack_kernel<<<1, 512, 0, stream>>>(kern, Bws);

    const int B = in_sizes[0] / C_DIM;                            // 131072
    crowds_wmma_kernel<<<B / TILE_B, THREADS, 0, stream>>>(x, Bws, mask, out);
}